// DocREModel_29137058136452
// MI455X (gfx1250) — compile-verified
//
#include <hip/hip_runtime.h>
#include <hip/hip_bf16.h>
#include <math.h>

typedef float v2f __attribute__((ext_vector_type(2)));
typedef float v8f __attribute__((ext_vector_type(8)));

constexpr int kBS = 4, kL = 1024, kH = 768, kHEADS = 12;
constexpr int kE = 32, kM = 8, kR = 992;
constexpr int kEMB = 768, kBLOCK = 64, kNC = 97;
constexpr int kTE = kBS * kE;                          // 128 entities
constexpr int kP  = kBS * kR;                          // 3968 pairs
constexpr int kG  = kEMB / kBLOCK;                     // 12 blocks
constexpr float kSMALL = -10000000000.0f;

static __device__ __forceinline__ v8f wmma4(v2f a, v2f b, v8f c) {
  // D = A(16x4 f32) * B(4x16 f32) + C(16x16 f32)
  return __builtin_amdgcn_wmma_f32_16x16x4_f32(false, a, false, b, (short)0, c,
                                               false, false);
}

// ---------------- K1: entity_embs = logsumexp over mentions ----------------
__global__ void k_entity_emb(const float* __restrict__ seq,
                             const int* __restrict__ epos,
                             float* __restrict__ eemb) {
  int e = blockIdx.x;
  int doc = e / kE;
  int pos[kM];
#pragma unroll
  for (int m = 0; m < kM; ++m) pos[m] = epos[e * kM + m];
  for (int h = threadIdx.x; h < kH; h += blockDim.x) {
    float mx = kSMALL;
#pragma unroll
    for (int m = 0; m < kM; ++m) {
      int p = pos[m];
      if (p >= 0) mx = fmaxf(mx, seq[((size_t)doc * kL + p) * kH + h]);
    }
    float s = 0.f;
#pragma unroll
    for (int m = 0; m < kM; ++m) {
      int p = pos[m];
      if (p >= 0) s += expf(seq[((size_t)doc * kL + p) * kH + h] - mx);
    }
    eemb[(size_t)e * kH + h] = mx + logf(s);
  }
}

// ---------------- K2: entity_attns = mean of mention attention rows --------
__global__ void k_entity_attn(const float* __restrict__ attn,
                              const int* __restrict__ epos,
                              float* __restrict__ eattn) {
  int bid = blockIdx.x;                 // e*HEADS + head
  int e = bid / kHEADS, head = bid % kHEADS;
  int doc = e / kE;
  int pos[kM]; int nv = 0;
#pragma unroll
  for (int m = 0; m < kM; ++m) {
    int p = epos[e * kM + m];
    pos[m] = p; nv += (p >= 0) ? 1 : 0;
  }
  float inv = 1.f / (float)nv;
  const float* base = attn + ((size_t)doc * kHEADS + head) * kL * kL;
  for (int l = threadIdx.x; l < kL; l += blockDim.x) {
    float s = 0.f;
#pragma unroll
    for (int m = 0; m < kM; ++m) {
      int p = pos[m];
      if (p >= 0) s += base[(size_t)p * kL + l];
    }
    eattn[((size_t)e * kHEADS + head) * kL + l] = s * inv;
  }
}

// ---------------- K3: rs = normalize(sum_heads ea[h]*ea[t]) ----------------
__global__ void k_rs(const float* __restrict__ eattn,
                     const int* __restrict__ hts,
                     float* __restrict__ rs) {
  int p = blockIdx.x;
  int doc = p / kR;
  int hg = hts[2 * p + 0] + doc * kE;
  int tg = hts[2 * p + 1] + doc * kE;
  const float* ha = eattn + (size_t)hg * kHEADS * kL;
  const float* ta = eattn + (size_t)tg * kHEADS * kL;
  float v[4];
  float tsum = 0.f;
#pragma unroll
  for (int i = 0; i < 4; ++i) {
    int l = threadIdx.x + i * 256;
    float s = 0.f;
#pragma unroll
    for (int hd = 0; hd < kHEADS; ++hd) s += ha[hd * kL + l] * ta[hd * kL + l];
    v[i] = s; tsum += s;
  }
  __shared__ float red[256];
  red[threadIdx.x] = tsum;
  __syncthreads();
  for (int s = 128; s > 0; s >>= 1) {
    if (threadIdx.x < s) red[threadIdx.x] += red[threadIdx.x + s];
    __syncthreads();
  }
  float inv = 1.f / red[0];
#pragma unroll
  for (int i = 0; i < 4; ++i) {
    int l = threadIdx.x + i * 256;
    rs[(size_t)p * kL + l] = v[i] * inv;
  }
}

// ---------------- K4: ctx = per-doc rs(992x1024) @ seq(1024x768) -----------
// grid.x = BS * (R/16) M-tiles, grid.y = H/128 N-groups, 256 thr = 8 waves,
// one 16x16 WMMA tile per wave.
__global__ void k_ctx(const float* __restrict__ rs,
                      const float* __restrict__ seq,
                      float* __restrict__ ctx) {
  int mt  = blockIdx.x % (kR / 16);
  int doc = blockIdx.x / (kR / 16);
  int lane = threadIdx.x & 31;
  int wave = threadIdx.x >> 5;
  int n0 = (blockIdx.y * 8 + wave) * 16;
  int rowbase = doc * kR + mt * 16;
  int mrow  = lane & 15;
  int khalf = (lane >> 4) << 1;                 // 0 or 2
  int ncol  = n0 + (lane & 15);
  const float* A = rs + (size_t)(rowbase + mrow) * kL;
  const float* B = seq + (size_t)doc * kL * kH;
  v8f acc = {};
#pragma unroll 4
  for (int k0 = 0; k0 < kL; k0 += 4) {
    int kb = k0 + khalf;
    v2f a = *(const v2f*)(A + kb);              // kb even -> 8B aligned
    v2f b;
    b[0] = B[(size_t)kb * kH + ncol];
    b[1] = B[(size_t)(kb + 1) * kH + ncol];
    acc = wmma4(a, b, acc);
  }
#pragma unroll
  for (int r = 0; r < 8; ++r) {
    int m = r + ((lane >> 4) << 3);
    ctx[(size_t)(rowbase + m) * kH + ncol] = acc[r];
  }
}

// ---------------- K5: hs/ts = tanh([emb|ctx](Px1536) @ W_head + b) ---------
// grid.x = P/16, grid.y = H/128, grid.z = 0(hs)/1(ts)
__global__ void k_head(const float* __restrict__ eemb,
                       const float* __restrict__ ctx,
                       const int* __restrict__ hts,
                       const float* __restrict__ Wh,
                       const float* __restrict__ bh,
                       float* __restrict__ outHs,
                       float* __restrict__ outTs) {
  int mt = blockIdx.x;
  int which = blockIdx.z;
  float* out = which ? outTs : outHs;
  __shared__ int s_ent[16];
  if (threadIdx.x < 16) {
    int p = mt * 16 + threadIdx.x;
    s_ent[threadIdx.x] = hts[2 * p + which] + (p / kR) * kE;
  }
  __syncthreads();
  int lane = threadIdx.x & 31;
  int wave = threadIdx.x >> 5;
  int n0 = (blockIdx.y * 8 + wave) * 16;
  int mrow  = lane & 15;
  int khalf = (lane >> 4) << 1;
  int ncol  = n0 + (lane & 15);
  int p = mt * 16 + mrow;
  const float* erow = eemb + (size_t)s_ent[mrow] * kH;
  const float* crow = ctx + (size_t)p * kH;
  v8f acc = {};
#pragma unroll 4
  for (int k0 = 0; k0 < 2 * kH; k0 += 4) {
    int kb = k0 + khalf;                        // even; kb,kb+1 same side of 768
    const float* arow = (kb < kH) ? (erow + kb) : (crow + (kb - kH));
    v2f a = *(const v2f*)arow;                  // 8B aligned
    v2f b;
    b[0] = Wh[(size_t)kb * kEMB + ncol];
    b[1] = Wh[(size_t)(kb + 1) * kEMB + ncol];
    acc = wmma4(a, b, acc);
  }
  float bias = bh[ncol];
#pragma unroll
  for (int r = 0; r < 8; ++r) {
    int m = r + ((lane >> 4) << 3);
    out[(size_t)(mt * 16 + m) * kH + ncol] = tanhf(acc[r] + bias);
  }
}

// ---------------- K6: logits = (hs ox_block ts) @ W_bl + b_bl --------------
// A(Px49152) built on the fly: A[p, g*4096+i*64+j] = hs[p,g*64+i]*ts[p,g*64+j]
// Loop nest g -> i -> j0 so the hs factor is hoisted across 16 WMMAs and the
// ts pair is one aligned b64 load. grid.x = P/16; waves 0..6 cover 97 cols.
__global__ void k_logits(const float* __restrict__ hs,
                         const float* __restrict__ ts,
                         const float* __restrict__ Wbl,
                         const float* __restrict__ bbl,
                         float* __restrict__ out) {
  int mt = blockIdx.x;
  int lane = threadIdx.x & 31;
  int wave = threadIdx.x >> 5;
  if (wave >= 7) return;                        // wave-uniform; 7*16=112 >= 97
  int c0 = wave * 16;
  int mrow  = lane & 15;
  int khalf = (lane >> 4) << 1;
  int ccol  = c0 + (lane & 15);
  int ccolr = (ccol < kNC) ? ccol : (kNC - 1);  // clamp loads for partial tile
  int p = mt * 16 + mrow;
  const float* hrow = hs + (size_t)p * kH;
  const float* trow = ts + (size_t)p * kH;
  v8f acc = {};
  for (int g = 0; g < kG; ++g) {
    const float* hg = hrow + g * kBLOCK;
    const float* tg = trow + g * kBLOCK;
    const float* Wg = Wbl + (size_t)g * kBLOCK * kBLOCK * kNC;
    for (int i = 0; i < kBLOCK; ++i) {
      float hv = hg[i];                         // invariant over 16 WMMAs
      const float* Wi = Wg + (size_t)i * kBLOCK * kNC;
#pragma unroll 4
      for (int j0 = 0; j0 < kBLOCK; j0 += 4) {
        int jj = j0 + khalf;                    // even -> 8B aligned
        v2f tv = *(const v2f*)(tg + jj);
        v2f a, b;
        a[0] = hv * tv[0];
        a[1] = hv * tv[1];
        b[0] = Wi[(size_t)jj * kNC + ccolr];
        b[1] = Wi[(size_t)(jj + 1) * kNC + ccolr];
        acc = wmma4(a, b, acc);
      }
    }
  }
  float bias = bbl[ccolr];
  if (ccol < kNC) {
#pragma unroll
    for (int r = 0; r < 8; ++r) {
      int m = r + ((lane >> 4) << 3);
      out[(size_t)(mt * 16 + m) * kNC + ccol] = acc[r] + bias;
    }
  }
}

extern "C" void kernel_launch(void* const* d_in, const int* in_sizes, int n_in,
                              void* d_out, int out_size, void* d_ws, size_t ws_size,
                              hipStream_t stream) {
  const float* seq  = (const float*)d_in[0];   // (4,1024,768)
  const float* attn = (const float*)d_in[1];   // (4,12,1024,1024)
  const int*   epos = (const int*)d_in[2];     // (128,8)
  const int*   hts  = (const int*)d_in[3];     // (3968,2)
  // d_in[4]=n_entities, d_in[5]=n_rels: uniform (E,R), folded into constants
  const float* Wh   = (const float*)d_in[6];   // (1536,768)
  const float* bh   = (const float*)d_in[7];   // (768,)
  const float* Wbl  = (const float*)d_in[8];   // (49152,97)
  const float* bbl  = (const float*)d_in[9];   // (97,)
  float* out = (float*)d_out;                  // (3968,97)

  float* ws    = (float*)d_ws;
  float* eemb  = ws;                                  // 128*768
  float* eattn = eemb  + (size_t)kTE * kH;            // 128*12*1024
  float* rs    = eattn + (size_t)kTE * kHEADS * kL;   // 3968*1024
  float* ctx   = rs    + (size_t)kP * kL;             // 3968*768
  float* hsb   = ctx   + (size_t)kP * kH;             // 3968*768
  float* tsb   = hsb   + (size_t)kP * kH;             // 3968*768
  (void)in_sizes; (void)n_in; (void)out_size; (void)ws_size;

  k_entity_emb <<<kTE,            256, 0, stream>>>(seq, epos, eemb);
  k_entity_attn<<<kTE * kHEADS,   256, 0, stream>>>(attn, epos, eattn);
  k_rs         <<<kP,             256, 0, stream>>>(eattn, hts, rs);
  k_ctx        <<<dim3(kBS * (kR / 16), kH / 128), 256, 0, stream>>>(rs, seq, ctx);
  k_head       <<<dim3(kP / 16, kH / 128, 2),      256, 0, stream>>>(eemb, ctx, hts, Wh, bh, hsb, tsb);
  k_logits     <<<kP / 16,        256, 0, stream>>>(hsb, tsb, Wbl, bbl, out);
}